// IDCT2DSpatialTransformLayer_52055003627866
// MI455X (gfx1250) — compile-verified
//
#include <hip/hip_runtime.h>

// IDCT2D: per (b,c) image  out = Wy * X * Wx^T,  X is 112x112 fp32.
// B*C = 4096 images, one workgroup each. fp32 WMMA (V_WMMA_F32_16X16X4_F32)
// with 2x2 register tiling per wave and a balanced static tile schedule.

typedef float v2f __attribute__((ext_vector_type(2)));
typedef float v4f __attribute__((ext_vector_type(4)));
typedef float v8f __attribute__((ext_vector_type(8)));

#define HW      112
#define LDP     116        // padded LDS row stride (floats): conflict-free b64 operand loads
#define TPB     256
#define NWAVES  (TPB/32)

// ---- one GEMM unit: RM x RN grid of 16x16 output tiles, K = 112 in steps of 4 ----
// A/B operands are fetched once per k-step and reused across the tile grid.
template<int RM, int RN>
__device__ __forceinline__ void unit_gemm(const float* __restrict__ As,
                                          const float* __restrict__ Bs,
                                          int mt0, int nt0, int lh, int koff,
                                          v8f (&acc)[RM][RN])
{
    const float* Ar[RM];
    const float* Br[RN];
#pragma unroll
    for (int r = 0; r < RM; ++r) Ar[r] = As + ((mt0 + r) * 16 + lh) * LDP + koff;
#pragma unroll
    for (int c = 0; c < RN; ++c) Br[c] = Bs + ((nt0 + c) * 16 + lh) * LDP + koff;
#pragma unroll
    for (int k0 = 0; k0 < HW; k0 += 4) {
        v2f a[RM], b[RN];
#pragma unroll
        for (int r = 0; r < RM; ++r) a[r] = *(const v2f*)(Ar[r] + k0);
#pragma unroll
        for (int c = 0; c < RN; ++c) b[c] = *(const v2f*)(Br[c] + k0);
#pragma unroll
        for (int r = 0; r < RM; ++r)
#pragma unroll
            for (int c = 0; c < RN; ++c)
                acc[r][c] = __builtin_amdgcn_wmma_f32_16x16x4_f32(
                    false, a[r], false, b[c], (short)0, acc[r][c], false, false);
    }
}

// store a D tile transposed into Tt[v][h] (two aligned b128 LDS stores)
__device__ __forceinline__ void store_Tt(float* __restrict__ Tts, const v8f& acc,
                                         int mt, int nt, int lh, int half)
{
    float* dst = Tts + (nt * 16 + lh) * LDP + mt * 16 + 8 * half;
    *(v4f*)(dst)     = (v4f){acc[0], acc[1], acc[2], acc[3]};
    *(v4f*)(dst + 4) = (v4f){acc[4], acc[5], acc[6], acc[7]};
}

// store a D tile to the global output (64B-per-half-wave coalesced)
__device__ __forceinline__ void store_out(float* __restrict__ og, const v8f& acc,
                                          int mt, int nt, int lh, int half)
{
    float* dst = og + (mt * 16 + 8 * half) * HW + nt * 16 + lh;
#pragma unroll
    for (int r = 0; r < 8; ++r) dst[r * HW] = acc[r];
}

template<int RM, int RN, bool STAGE2>
__device__ __forceinline__ void run_unit(const float* __restrict__ As,
                                         const float* __restrict__ Bs,
                                         float* __restrict__ Tts,
                                         float* __restrict__ og,
                                         int mt0, int nt0, int lh, int half, int koff)
{
    v8f acc[RM][RN] = {};
    unit_gemm<RM, RN>(As, Bs, mt0, nt0, lh, koff, acc);
#pragma unroll
    for (int r = 0; r < RM; ++r)
#pragma unroll
        for (int c = 0; c < RN; ++c) {
            if (STAGE2) store_out(og,  acc[r][c], mt0 + r, nt0 + c, lh, half);
            else        store_Tt(Tts, acc[r][c], mt0 + r, nt0 + c, lh, half);
        }
}

template<bool STAGE2>
__device__ __forceinline__ void run_stage(const float* __restrict__ As,
                                          const float* __restrict__ Bs,
                                          float* __restrict__ Tts,
                                          float* __restrict__ og,
                                          int wv, int lh, int half, int koff)
{
    // Balanced static schedule over the 7x7 tile grid:
    //   8 full 2x2 units, 8 two-tile units (five 2x1, three 1x2), 1 single (wave 7).
    int fm, fn, pm, pn, pt; // pt: 0 => 2x1 pair, 1 => 1x2 pair
    switch (wv) {
        case 0:  fm = 0; fn = 0; pm = 0; pn = 6; pt = 0; break;
        case 1:  fm = 0; fn = 2; pm = 2; pn = 6; pt = 0; break;
        case 2:  fm = 0; fn = 4; pm = 4; pn = 6; pt = 0; break;
        case 3:  fm = 2; fn = 0; pm = 4; pn = 4; pt = 0; break;
        case 4:  fm = 2; fn = 2; pm = 4; pn = 5; pt = 0; break;
        case 5:  fm = 2; fn = 4; pm = 6; pn = 0; pt = 1; break;
        case 6:  fm = 4; fn = 0; pm = 6; pn = 2; pt = 1; break;
        default: fm = 4; fn = 2; pm = 6; pn = 4; pt = 1; break;
    }

    run_unit<2, 2, STAGE2>(As, Bs, Tts, og, fm, fn, lh, half, koff);
    if (pt == 0) run_unit<2, 1, STAGE2>(As, Bs, Tts, og, pm, pn, lh, half, koff);
    else         run_unit<1, 2, STAGE2>(As, Bs, Tts, og, pm, pn, lh, half, koff);
    if (wv == 7) run_unit<1, 1, STAGE2>(As, Bs, Tts, og, 6, 6, lh, half, koff);
}

__global__ __launch_bounds__(TPB)
void idct2d_wmma_f32(const float* __restrict__ x,
                     const float* __restrict__ Wx,
                     const float* __restrict__ Wy,
                     float* __restrict__ out)
{
    extern __shared__ float smem[];
    float* Xs  = smem;                  // X  [112][116] row-major (h, j)
    float* Wxs = Xs  + HW * LDP;        // Wx [112][116] row-major (v, j)
    float* Wys = Wxs + HW * LDP;        // Wy [112][116] row-major (m, k)
    float* Tts = Wys + HW * LDP;        // T^T [112][116]: Tt[v][h] = T[h][v]

    const int bc = blockIdx.x;
    const float* xg = x   + (size_t)bc * (HW * HW);
    float*       og = out + (size_t)bc * (HW * HW);

    // ---------------- load phase: X, Wx, Wy -> LDS (float4, coalesced) ----------------
    const int tid = threadIdx.x;
    for (int idx = tid; idx < HW * (HW / 4); idx += TPB) {
        const int row = idx / (HW / 4);
        const int c4  = (idx % (HW / 4)) * 4;
        v4f vx = *(const v4f*)(xg + row * HW + c4);
        v4f va = *(const v4f*)(Wx + row * HW + c4);
        v4f vb = *(const v4f*)(Wy + row * HW + c4);
        *(v4f*)(Xs  + row * LDP + c4) = vx;
        *(v4f*)(Wxs + row * LDP + c4) = va;
        *(v4f*)(Wys + row * LDP + c4) = vb;
    }
    __syncthreads();

    const int lane = tid & 31;
    const int wv   = tid >> 5;
    const int lh   = lane & 15;      // N / M row within tile for this lane
    const int half = lane >> 4;      // K sub-phase: lanes 16..31 carry K+2,K+3
    const int koff = 2 * half;

    // stage 1: T = X * Wx^T  (A[m][k] = X[h][j], B[k][n] = Wx[v][j], both j-contiguous)
    run_stage<false>(Xs, Wxs, Tts, og, wv, lh, half, koff);
    __syncthreads();

    // stage 2: O = Wy * T    (A[m][k] = Wy[m][k], B[k][n] = Tt[n][k], both k-contiguous)
    run_stage<true>(Wys, Tts, Tts, og, wv, lh, half, koff);
}

extern "C" void kernel_launch(void* const* d_in, const int* in_sizes, int n_in,
                              void* d_out, int out_size, void* d_ws, size_t ws_size,
                              hipStream_t stream) {
    const float* x  = (const float*)d_in[0];
    const float* Wx = (const float*)d_in[1];
    const float* Wy = (const float*)d_in[2];
    float* out = (float*)d_out;

    const int n_images = 32 * 128;                       // B * C
    const size_t shmem = 4u * HW * LDP * sizeof(float);  // ~203 KB of the 320 KB WGP LDS

    idct2d_wmma_f32<<<dim3(n_images), dim3(TPB), shmem, stream>>>(x, Wx, Wy, out);
}